// DAGNN_11897059410771
// MI455X (gfx1250) — compile-verified
//
#include <hip/hip_runtime.h>
#include <hip/hip_bf16.h>

// ---------------------------------------------------------------------------
// AttnConv: per-edge attention score, scatter-softmax over dst, weighted
// scatter-add.  N=100000 nodes, E=1600000 edges, D=64.
//
// Pass 1: alpha_q = h_attn_q @ Wq, alpha_k = h @ Wk   (WMMA f32 16x16x4)
// Pass 2: scores[e] = alpha_q[dst] + alpha_k[src] + b ; segment max (u32 keys)
// Pass 3: e[e] = exp(scores - max[dst]) ; segment sum (atomic f32)
// Pass 4: out[dst] += h[src] * e[e]/(sum[dst]+1e-16)  (b128 gather, f32 atomics)
// ---------------------------------------------------------------------------

typedef __attribute__((ext_vector_type(2))) float v2f;
typedef __attribute__((ext_vector_type(8))) float v8f;

#define D_DIM 64

// Monotone order-preserving float<->uint encoding for exact atomic max.
__device__ __forceinline__ unsigned f2key(float f) {
    unsigned u = __float_as_uint(f);
    return (u & 0x80000000u) ? ~u : (u | 0x80000000u);
}
__device__ __forceinline__ float key2f(unsigned k) {
    unsigned u = (k & 0x80000000u) ? (k ^ 0x80000000u) : ~k;
    return __uint_as_float(u);
}

// ---------------------------------------------------------------------------
// Pass 0: zero out[N*64] (float4 stores), seg_sum[N], segmax keys[N].
// ---------------------------------------------------------------------------
__global__ void init_kernel(float4* __restrict__ out4, float* __restrict__ seg_sum,
                            unsigned* __restrict__ segmax, int n) {
    int i = blockIdx.x * blockDim.x + threadIdx.x;
    if (i < n * (D_DIM / 4)) out4[i] = make_float4(0.f, 0.f, 0.f, 0.f);
    if (i < n) { seg_sum[i] = 0.0f; segmax[i] = 0u; }
}

// ---------------------------------------------------------------------------
// Pass 1: per-node dot products via V_WMMA_F32_16X16X4_F32.
// One wave handles 16 rows. blockIdx.y = 0 -> (h_attn_q, W[0:64])  -> alpha_q
//                           blockIdx.y = 1 -> (h,        W[64:128])-> alpha_k
// A fragment (16x4 f32): VGPR r, lane half h holds element (M = lane%16,
// K = k0 + r + 2*h).  B fragment (4x16): same K mapping, value broadcast
// across the 16 N-columns => every column of D equals the dot product.
// D (16x16 f32): VGPR r, lane 0 -> (M=r, N=0); lane 16 -> (M=8+r, N=0).
// ---------------------------------------------------------------------------
__global__ __launch_bounds__(256) void alpha_wmma_kernel(
    const float* __restrict__ Xq, const float* __restrict__ Xk,
    const float* __restrict__ W, float* __restrict__ alpha_q,
    float* __restrict__ alpha_k, int n) {
    const int wave = (blockIdx.x * blockDim.x + threadIdx.x) >> 5;
    const int lane = threadIdx.x & 31;
    const int base = wave * 16;
    if (base >= n) return;  // uniform per wave: EXEC stays all-ones for WMMA

    const float* X = blockIdx.y ? Xk : Xq;
    const float* w = W + (blockIdx.y ? D_DIM : 0);
    float* outp    = blockIdx.y ? alpha_k : alpha_q;

    const int half = lane >> 4;   // K-half selector
    const int m    = lane & 15;   // row within tile
    int row = base + m;
    if (row >= n) row = n - 1;    // clamped read, guarded write
    const float* xrow = X + (size_t)row * D_DIM;

    v8f c = {};
#pragma unroll
    for (int k0 = 0; k0 < D_DIM; k0 += 4) {
        const int ka = k0 + 2 * half;
        v2f a, b;
        a.x = xrow[ka + 0];
        a.y = xrow[ka + 1];
        b.x = w[ka + 0];
        b.y = w[ka + 1];
        c = __builtin_amdgcn_wmma_f32_16x16x4_f32(
            /*neg_a=*/false, a, /*neg_b=*/false, b,
            /*c_mod=*/(short)0, c, /*reuse_a=*/false, /*reuse_b=*/false);
    }

    if (m == 0) {  // lanes 0 and 16 hold column N=0 of D
        const int moff = half * 8;
#pragma unroll
        for (int r = 0; r < 8; ++r) {
            int idx = base + moff + r;
            if (idx < n) outp[idx] = c[r];
        }
    }
}

// ---------------------------------------------------------------------------
// Pass 2: per-edge score + exact segment max via u32 atomic max.
// ---------------------------------------------------------------------------
__global__ void score_kernel(const int* __restrict__ src, const int* __restrict__ dst,
                             const float* __restrict__ alpha_q,
                             const float* __restrict__ alpha_k,
                             const float* __restrict__ b_attn,
                             float* __restrict__ scores, unsigned* __restrict__ segmax,
                             int e_cnt) {
    int e = blockIdx.x * blockDim.x + threadIdx.x;
    if (e >= e_cnt) return;
    int di = dst[e], si = src[e];
    float s = alpha_q[di] + alpha_k[si] + b_attn[0];
    scores[e] = s;
    atomicMax(&segmax[di], f2key(s));
}

// ---------------------------------------------------------------------------
// Pass 3: e = exp(score - max[dst]) (in place) + segment sum.
// ---------------------------------------------------------------------------
__global__ void exp_kernel(const int* __restrict__ dst, const unsigned* __restrict__ segmax,
                           float* __restrict__ scores, float* __restrict__ seg_sum,
                           int e_cnt) {
    int e = blockIdx.x * blockDim.x + threadIdx.x;
    if (e >= e_cnt) return;
    int di = dst[e];
    float v = __expf(scores[e] - key2f(segmax[di]));
    scores[e] = v;
    atomicAdd(&seg_sum[di], v);
}

// ---------------------------------------------------------------------------
// Pass 4: out[dst] += h[src] * a.  16 threads per edge, 4 elems per thread:
// global_load_b128 gather of h, 4 f32 atomics into L2-resident out.
// ---------------------------------------------------------------------------
__global__ void scatter_kernel(const int* __restrict__ src, const int* __restrict__ dst,
                               const float* __restrict__ h, const float* __restrict__ scores,
                               const float* __restrict__ seg_sum, float* __restrict__ out,
                               int e_cnt) {
    int t = blockIdx.x * blockDim.x + threadIdx.x;
    int e = t >> 4;
    if (e >= e_cnt) return;
    int d = (t & 15) << 2;
    int di = dst[e], si = src[e];
    float a = scores[e] / (seg_sum[di] + 1e-16f);
    float4 hv = *(const float4*)(h + (size_t)si * D_DIM + d);
    float* op = out + (size_t)di * D_DIM + d;
    atomicAdd(op + 0, hv.x * a);
    atomicAdd(op + 1, hv.y * a);
    atomicAdd(op + 2, hv.z * a);
    atomicAdd(op + 3, hv.w * a);
}

// ---------------------------------------------------------------------------
// Host-side launch.
// Inputs (setup_inputs order): h [N*64] f32, h_attn_q [N*64] f32,
// W_attn [128] f32, b_attn [1] f32, edge_index [2*E] int32 (src row, dst row).
// ---------------------------------------------------------------------------
extern "C" void kernel_launch(void* const* d_in, const int* in_sizes, int n_in,
                              void* d_out, int out_size, void* d_ws, size_t ws_size,
                              hipStream_t stream) {
    const float* h        = (const float*)d_in[0];
    const float* h_attn_q = (const float*)d_in[1];
    const float* W_attn   = (const float*)d_in[2];
    const float* b_attn   = (const float*)d_in[3];
    const int*   eidx     = (const int*)d_in[4];

    const int N = in_sizes[0] / D_DIM;
    const int E = in_sizes[4] / 2;
    const int* src = eidx;       // edge_index[0]
    const int* dst = eidx + E;   // edge_index[1]

    // Workspace layout: alpha_q[N] | alpha_k[N] | seg_sum[N] | segmax[N] | scores[E]
    float*    alpha_q = (float*)d_ws;
    float*    alpha_k = alpha_q + N;
    float*    seg_sum = alpha_k + N;
    unsigned* segmax  = (unsigned*)(seg_sum + N);
    float*    scores  = (float*)(segmax + N);

    float* out = (float*)d_out;

    const int TB = 256;

    // Pass 0: zero accumulators (d_out is poisoned by the harness).
    {
        int total4 = N * (D_DIM / 4);
        init_kernel<<<(total4 + TB - 1) / TB, TB, 0, stream>>>((float4*)out, seg_sum,
                                                               segmax, N);
    }

    // Pass 1: WMMA projections (grid.y selects q/k).
    {
        int waves  = (N + 15) / 16;
        int blocks = (waves + (TB / 32) - 1) / (TB / 32);
        dim3 grid(blocks, 2);
        alpha_wmma_kernel<<<grid, TB, 0, stream>>>(h_attn_q, h, W_attn,
                                                   alpha_q, alpha_k, N);
    }

    // Pass 2: scores + segment max.
    score_kernel<<<(E + TB - 1) / TB, TB, 0, stream>>>(src, dst, alpha_q, alpha_k,
                                                       b_attn, scores, segmax, E);

    // Pass 3: exp + segment sum.
    exp_kernel<<<(E + TB - 1) / TB, TB, 0, stream>>>(dst, segmax, scores, seg_sum, E);

    // Pass 4: normalized weighted scatter-add.
    {
        long long threads = (long long)E * 16;
        int blocks = (int)((threads + TB - 1) / TB);
        scatter_kernel<<<blocks, TB, 0, stream>>>(src, dst, h, scores, seg_sum, out, E);
    }
}